// GCNNet_7198365188473
// MI455X (gfx1250) — compile-verified
//
#include <hip/hip_runtime.h>

typedef __attribute__((ext_vector_type(2))) float v2f;
typedef __attribute__((ext_vector_type(8))) float v8f;

#define NODES 50000
#define CH 512
#define COUT 40

// ---------------- degree / normalization ----------------
__global__ void k_deg_init(float* deg, int n) {
  int i = blockIdx.x * blockDim.x + threadIdx.x;
  if (i < n) deg[i] = 1.0f;  // self loop
}

__global__ void k_deg_edges(const int* __restrict__ dst, float* deg, int E) {
  int e = blockIdx.x * blockDim.x + threadIdx.x;
  if (e < E) atomicAdd(&deg[dst[e]], 1.0f);
}

__global__ void k_dinv(float* deg, int n) {
  int i = blockIdx.x * blockDim.x + threadIdx.x;
  if (i < n) deg[i] = rsqrtf(deg[i]);  // deg >= 1 always
}

// ---------------- weight transposes (once per launch, tiny) ----------------
// W1t[n*512 + k] = W1[k*512 + n]  (512x512)
__global__ void k_transpose_w1(const float* __restrict__ B, float* __restrict__ Bt) {
  int idx = blockIdx.x * 256 + threadIdx.x;   // exactly 512*512 threads
  int k = idx & 511;
  int n = idx >> 9;
  Bt[idx] = B[(size_t)k * 512 + n];
}

// W2t: 48x512, rows 40..47 zero-padded. W2t[n*512 + k] = (n<40) ? W2[k*40+n] : 0
__global__ void k_transpose_w2(const float* __restrict__ B, float* __restrict__ Bt) {
  int idx = blockIdx.x * 256 + threadIdx.x;   // exactly 48*512 threads
  int k = idx & 511;
  int n = idx >> 9;
  Bt[idx] = (n < COUT) ? B[(size_t)k * COUT + n] : 0.0f;
}

// ---------------- GEMM1: [M,512] x [512,512], full f32 WMMA ----------------
// One wave -> 16(M) x 64(N) tile, 4 accumulators. Block = 8 waves = 16 x 512.
// B pre-transposed so every fragment is one contiguous b64 load.
__global__ void __launch_bounds__(256) k_gemm_512x512(
    const float* __restrict__ A, const float* __restrict__ Bt, float* __restrict__ C) {
  const int K = 512, N = 512;
  int lane  = threadIdx.x & 31;
  int wave  = threadIdx.x >> 5;
  int mtile = blockIdx.x;              // 0..3124 (50000/16 exact)
  int n0    = wave << 6;               // 64 cols per wave
  int l15   = lane & 15;
  int khalf = lane >> 4;               // 0 or 1

  const float* arow = A + (size_t)(mtile * 16 + l15) * K;
  const float* bt0  = Bt + (size_t)(n0 + l15) * K;
  const float* bt1  = bt0 + 16 * K;
  const float* bt2  = bt0 + 32 * K;
  const float* bt3  = bt0 + 48 * K;

  v8f acc0 = {}, acc1 = {}, acc2 = {}, acc3 = {};
  for (int k = 0; k < K; k += 4) {
    int ka = k + (khalf << 1);
    v2f a   = *(const v2f*)(arow + ka);
    v2f bb0 = *(const v2f*)(bt0 + ka);
    v2f bb1 = *(const v2f*)(bt1 + ka);
    v2f bb2 = *(const v2f*)(bt2 + ka);
    v2f bb3 = *(const v2f*)(bt3 + ka);
    acc0 = __builtin_amdgcn_wmma_f32_16x16x4_f32(false, a, false, bb0, (short)0, acc0, false, false);
    acc1 = __builtin_amdgcn_wmma_f32_16x16x4_f32(false, a, false, bb1, (short)0, acc1, false, false);
    acc2 = __builtin_amdgcn_wmma_f32_16x16x4_f32(false, a, false, bb2, (short)0, acc2, false, false);
    acc3 = __builtin_amdgcn_wmma_f32_16x16x4_f32(false, a, false, bb3, (short)0, acc3, false, false);
  }

  int rbase = mtile * 16 + (khalf << 3);
  float* crow = C + (size_t)rbase * N + n0 + l15;
#pragma unroll
  for (int v = 0; v < 8; v++) {
    crow[0]  = acc0[v];
    crow[16] = acc1[v];
    crow[32] = acc2[v];
    crow[48] = acc3[v];
    crow += N;
  }
}

// ---------------- GEMM2: [M,512] x [512,40] via zero-padded 48x512 Bt ----------------
__global__ void __launch_bounds__(256) k_gemm_512x40(
    const float* __restrict__ A, const float* __restrict__ Bt, float* __restrict__ C,
    int Mtiles) {
  const int K = 512, NO = COUT;
  int lane  = threadIdx.x & 31;
  int wave  = threadIdx.x >> 5;
  int mtile = blockIdx.x * 8 + wave;
  if (mtile >= Mtiles) return;  // wave-uniform: EXEC stays all-1s for live waves
  int l15   = lane & 15;
  int khalf = lane >> 4;
  bool has2 = (l15 < 8);        // third sub-tile column = l15+32 < 40

  const float* arow = A + (size_t)(mtile * 16 + l15) * K;
  const float* bt0  = Bt + (size_t)l15 * K;
  const float* bt1  = bt0 + 16 * K;
  const float* bt2  = bt0 + 32 * K;     // rows >= 40 are zero-padded

  v8f acc0 = {}, acc1 = {}, acc2 = {};
  for (int k = 0; k < K; k += 4) {
    int ka = k + (khalf << 1);
    v2f a   = *(const v2f*)(arow + ka);
    v2f bb0 = *(const v2f*)(bt0 + ka);
    v2f bb1 = *(const v2f*)(bt1 + ka);
    v2f bb2 = *(const v2f*)(bt2 + ka);
    acc0 = __builtin_amdgcn_wmma_f32_16x16x4_f32(false, a, false, bb0, (short)0, acc0, false, false);
    acc1 = __builtin_amdgcn_wmma_f32_16x16x4_f32(false, a, false, bb1, (short)0, acc1, false, false);
    acc2 = __builtin_amdgcn_wmma_f32_16x16x4_f32(false, a, false, bb2, (short)0, acc2, false, false);
  }

  int rbase = mtile * 16 + (khalf << 3);
  float* crow = C + (size_t)rbase * NO + l15;
#pragma unroll
  for (int v = 0; v < 8; v++) {
    crow[0]  = acc0[v];
    crow[16] = acc1[v];
    if (has2) crow[32] = acc2[v];
    crow += NO;
  }
}

// ---------------- aggregation (layer 1, 512 feats) ----------------
__global__ void k_self_512(const float* __restrict__ h, const float* __restrict__ dinv,
                           float* __restrict__ out) {
  int idx = blockIdx.x * 256 + threadIdx.x;   // exactly NODES*512 threads
  int node = idx >> 9;
  float di = dinv[node];
  out[idx] = h[idx] * di * di;
}

__global__ void __launch_bounds__(128) k_agg_edge_512(
    const float* __restrict__ h, const int* __restrict__ src, const int* __restrict__ dst,
    const float* __restrict__ dinv, float* __restrict__ out, int E) {
  int e = blockIdx.x;
  if (e >= E) return;
  int s = src[e], d = dst[e];
  float nrm = dinv[s] * dinv[d];
  const float4* hs = (const float4*)(h + (size_t)s * CH) + threadIdx.x;
  float* od = out + (size_t)d * CH + threadIdx.x * 4;
  float4 v = *hs;
  atomicAdd(od + 0, v.x * nrm);
  atomicAdd(od + 1, v.y * nrm);
  atomicAdd(od + 2, v.z * nrm);
  atomicAdd(od + 3, v.w * nrm);
}

__global__ void k_bias_relu_512(float* __restrict__ a, const float* __restrict__ b) {
  int idx = blockIdx.x * 256 + threadIdx.x;
  float v = a[idx] + b[idx & 511];
  a[idx] = v > 0.0f ? v : 0.0f;
}

// ---------------- aggregation (layer 2, 40 feats) ----------------
__global__ void k_self_40(const float* __restrict__ h, const float* __restrict__ dinv,
                          float* __restrict__ out, int total) {
  int idx = blockIdx.x * 256 + threadIdx.x;
  if (idx >= total) return;
  int node = idx / COUT;
  float di = dinv[node];
  out[idx] = h[idx] * di * di;
}

__global__ void __launch_bounds__(256) k_agg_edge_40(
    const float* __restrict__ h, const int* __restrict__ src, const int* __restrict__ dst,
    const float* __restrict__ dinv, float* __restrict__ out, int E) {
  int e = blockIdx.x * 8 + (threadIdx.x >> 5);
  if (e >= E) return;
  int lane = threadIdx.x & 31;
  int s = src[e], d = dst[e];
  float nrm = dinv[s] * dinv[d];
  const float* hs = h + (size_t)s * COUT;
  float* od = out + (size_t)d * COUT;
  atomicAdd(od + lane, hs[lane] * nrm);
  if (lane < 8) atomicAdd(od + 32 + lane, hs[32 + lane] * nrm);
}

// ---------------- bias + log_softmax over 40 classes (wave per node) ----------------
__global__ void __launch_bounds__(256) k_logsoftmax_40(
    const float* __restrict__ agg, const float* __restrict__ bias,
    float* __restrict__ out, int Nn) {
  int node = blockIdx.x * 8 + (threadIdx.x >> 5);
  if (node >= Nn) return;
  int lane = threadIdx.x & 31;
  const float* a = agg + (size_t)node * COUT;
  float v0 = a[lane] + bias[lane];
  bool hi = lane < 8;
  float v1 = hi ? (a[32 + lane] + bias[32 + lane]) : -__builtin_inff();
  float m = fmaxf(v0, v1);
#pragma unroll
  for (int off = 16; off; off >>= 1) m = fmaxf(m, __shfl_xor(m, off, 32));
  float s = expf(v0 - m) + (hi ? expf(v1 - m) : 0.0f);
#pragma unroll
  for (int off = 16; off; off >>= 1) s += __shfl_xor(s, off, 32);
  float lse = m + logf(s);
  float* o = out + (size_t)node * COUT;
  o[lane] = v0 - lse;
  if (hi) o[32 + lane] = v1 - lse;
}

// ---------------- launcher ----------------
extern "C" void kernel_launch(void* const* d_in, const int* in_sizes, int n_in,
                              void* d_out, int out_size, void* d_ws, size_t ws_size,
                              hipStream_t stream) {
  const float* x  = (const float*)d_in[0];
  const int*   ei = (const int*)d_in[1];   // [2, E] int32
  const float* W1 = (const float*)d_in[2];
  const float* b1 = (const float*)d_in[3];
  const float* W2 = (const float*)d_in[4];
  const float* b2 = (const float*)d_in[5];
  float* out = (float*)d_out;

  const int Nn = NODES;
  const int E  = in_sizes[1] / 2;
  const int* src = ei;
  const int* dst = ei + E;

  float* ws   = (float*)d_ws;
  float* deg  = ws;                              // 50000 (becomes dinv in place)
  float* h1   = ws + 51200;                      // NODES*512
  float* agg1 = h1 + (size_t)Nn * CH;            // NODES*512
  float* h2   = agg1 + (size_t)Nn * CH;          // NODES*40
  float* agg2 = h2 + (size_t)Nn * COUT;          // NODES*40
  float* W1t  = agg2 + (size_t)Nn * COUT;        // 512*512
  float* W2t  = W1t + 512 * 512;                 // 48*512 (zero-padded)

  // normalization
  k_deg_init<<<(Nn + 255) / 256, 256, 0, stream>>>(deg, Nn);
  k_deg_edges<<<(E + 255) / 256, 256, 0, stream>>>(dst, deg, E);
  k_dinv<<<(Nn + 255) / 256, 256, 0, stream>>>(deg, Nn);

  // weight transposes
  k_transpose_w1<<<(512 * 512) / 256, 256, 0, stream>>>(W1, W1t);
  k_transpose_w2<<<(48 * 512) / 256, 256, 0, stream>>>(W2, W2t);

  // layer 1
  k_gemm_512x512<<<Nn / 16, 256, 0, stream>>>(x, W1t, h1);
  k_self_512<<<(Nn * CH) / 256, 256, 0, stream>>>(h1, deg, agg1);
  k_agg_edge_512<<<E, 128, 0, stream>>>(h1, src, dst, deg, agg1, E);
  k_bias_relu_512<<<(Nn * CH) / 256, 256, 0, stream>>>(agg1, b1);

  // layer 2
  int Mtiles = Nn / 16;
  k_gemm_512x40<<<(Mtiles + 7) / 8, 256, 0, stream>>>(agg1, W2t, h2, Mtiles);
  int tot2 = Nn * COUT;
  k_self_40<<<(tot2 + 255) / 256, 256, 0, stream>>>(h2, deg, agg2, tot2);
  k_agg_edge_40<<<(E + 7) / 8, 256, 0, stream>>>(h2, src, dst, deg, agg2, E);
  k_logsoftmax_40<<<(Nn + 7) / 8, 256, 0, stream>>>(agg2, b2, out, Nn);
}